// Conv2DRaw_60516089200806
// MI455X (gfx1250) — compile-verified
//
#include <hip/hip_runtime.h>

typedef float v2f __attribute__((ext_vector_type(2)));
typedef float v8f __attribute__((ext_vector_type(8)));

// CDNA5 async global->LDS staging (ASYNCcnt-tracked), with safe fallback if the
// toolchain doesn't declare the builtins.
#if defined(__HIP_DEVICE_COMPILE__) && \
    __has_builtin(__builtin_amdgcn_global_load_async_to_lds_b32) && \
    __has_builtin(__builtin_amdgcn_s_wait_asynccnt)
#define USE_ASYNC_LDS 1
#else
#define USE_ASYNC_LDS 0
#endif

#if USE_ASYNC_LDS
// Builtin signature (from hipcc diagnostics): (addrspace(1) int*, addrspace(3) int*, imm, imm)
#define GASYNC_B32(gsrc, ldst)                                         \
    __builtin_amdgcn_global_load_async_to_lds_b32(                     \
        (__attribute__((address_space(1))) int*)(void*)(gsrc),         \
        (__attribute__((address_space(3))) int*)(ldst), 0, 0)
#define WAIT_ASYNC() __builtin_amdgcn_s_wait_asynccnt(0)
#else
#define WAIT_ASYNC() ((void)0)
#endif

namespace {
constexpr int C_IN   = 128;
constexpr int H_IMG  = 56;
constexpr int W_IMG  = 56;
constexpr int HW     = H_IMG * W_IMG;   // 3136
constexpr int K_OUT  = 256;
constexpr int K_RED  = C_IN * 9;        // 1152, ordered as k = rs*128 + c
constexpr int N_IMGS = 32;

constexpr int TILE_M   = 64;    // output channels per block (WMMA M axis)
constexpr int TILE_N   = 128;   // pixels per block        (WMMA N axis)
constexpr int K_STEP   = 16;    // channels staged per iteration
constexpr int A_STRIDE = 18;    // padded A tile row stride (even -> b64 frags)
constexpr int B_STRIDE = 132;   // padded B tile row stride
constexpr int N_ITER   = 9 * (C_IN / K_STEP);   // 72 K-slices
}

__global__ __launch_bounds__(256)
void conv3x3_wmma_f32(const float* __restrict__ x,
                      const float* __restrict__ wgt,
                      const float* __restrict__ bias,
                      float* __restrict__ out)
{
    // Ping-pong tiles: async fill of slice it+1 overlaps WMMA on slice it.
    __shared__ float a_lds[2][TILE_M * A_STRIDE];   // 2 x 4.5 KB
    __shared__ float b_lds[2][K_STEP * B_STRIDE];   // 2 x 8.25 KB

    const int tid    = threadIdx.x;
    const int m_tile = blockIdx.y * TILE_M;      // channel tile base
    const int p_tile = blockIdx.x * TILE_N;      // pixel tile base

    // ---- A (weight) loader: 64 rows x 16 c; src index = o*1152 + c*9 + rs ----
    const int a_m = tid & 63;
    const int a_k = (tid >> 6) << 2;             // 4 channels per thread
    const float* a_base = wgt + (size_t)(m_tile + a_m) * K_RED + (size_t)a_k * 9;

    // ---- B (im2col) loader: 16 c x 128 pixels; 8 channels per thread ----
    const int b_p  = tid & 127;
    const int b_k0 = (tid >> 7) << 3;
    const int gp    = p_tile + b_p;
    const int n_img = gp / HW;
    const int phw   = gp - n_img * HW;
    const int oh    = phw / W_IMG;
    const int ow    = phw - oh * W_IMG;
    const float* x_img = x + (size_t)n_img * C_IN * HW + (size_t)b_k0 * HW + phw;

    // ---- wave / lane mapping: 8 waves -> 2 (M) x 4 (N) grid of 32x32 tiles ----
    const int lane = tid & 31;
    const int wv   = tid >> 5;
    const int wm   = (wv & 1) * 32;
    const int wn   = (wv >> 1) * 32;
    const int half = lane >> 4;
    const int l16  = lane & 15;

    const int a_off = a_m * A_STRIDE + a_k;
    const int b_off = b_k0 * B_STRIDE + b_p;

    v8f acc[2][2];
    #pragma unroll
    for (int i = 0; i < 2; ++i)
        #pragma unroll
        for (int j = 0; j < 2; ++j)
            acc[i][j] = (v8f){0.f, 0.f, 0.f, 0.f, 0.f, 0.f, 0.f, 0.f};

    // Stage K-slice `it` (k = rs*128 + c ordering) into buffer `buf`.
    // rs/ct/r/s math is uniform -> SALU; halo predicate is 3-4 VALU ops.
    auto stage = [&](int it, int buf) {
        const int rs = it >> 3;
        const int ct = (it & 7) << 4;
        const int r  = rs / 3;
        const int s  = rs - r * 3;
        const int hi = oh + r - 1;
        const int wi = ow + s - 1;
        const bool valid = ((unsigned)hi < (unsigned)H_IMG) &
                           ((unsigned)wi < (unsigned)W_IMG);
        const float* a_src = a_base + rs + (size_t)ct * 9;
        const float* b_src = x_img + ((r - 1) * W_IMG + (s - 1)) + (size_t)ct * HW;
        float* as = &a_lds[buf][a_off];
        float* bs = &b_lds[buf][b_off];
#if USE_ASYNC_LDS
        #pragma unroll
        for (int j = 0; j < 4; ++j)
            GASYNC_B32(a_src + (size_t)j * 9, as + j);
        if (valid) {
            #pragma unroll
            for (int i = 0; i < 8; ++i)
                GASYNC_B32(b_src + (size_t)i * HW, bs + i * B_STRIDE);
        } else {
            // thread-owned slots, disjoint from all async targets this slice
            #pragma unroll
            for (int i = 0; i < 8; ++i)
                bs[i * B_STRIDE] = 0.0f;
        }
#else
        float av[4];
        #pragma unroll
        for (int j = 0; j < 4; ++j)
            av[j] = a_src[(size_t)j * 9];
        float bv[8];
        #pragma unroll
        for (int i = 0; i < 8; ++i)
            bv[i] = valid ? b_src[(size_t)i * HW] : 0.0f;
        #pragma unroll
        for (int j = 0; j < 4; ++j)
            as[j] = av[j];
        #pragma unroll
        for (int i = 0; i < 8; ++i)
            bs[i * B_STRIDE] = bv[i];
#endif
    };

    // 4 k-steps of 4 -> 16 f32 WMMA per slice per wave.
    auto compute = [&](int buf) {
        #pragma unroll
        for (int kk = 0; kk < K_STEP; kk += 4) {
            v2f a0 = *(const v2f*)&a_lds[buf][(wm +      l16) * A_STRIDE + kk + half * 2];
            v2f a1 = *(const v2f*)&a_lds[buf][(wm + 16 + l16) * A_STRIDE + kk + half * 2];
            v2f b0, b1;
            b0[0] = b_lds[buf][(kk + half * 2 + 0) * B_STRIDE + wn + l16];
            b0[1] = b_lds[buf][(kk + half * 2 + 1) * B_STRIDE + wn + l16];
            b1[0] = b_lds[buf][(kk + half * 2 + 0) * B_STRIDE + wn + 16 + l16];
            b1[1] = b_lds[buf][(kk + half * 2 + 1) * B_STRIDE + wn + 16 + l16];

            acc[0][0] = __builtin_amdgcn_wmma_f32_16x16x4_f32(false, a0, false, b0,
                            (short)0, acc[0][0], false, false);
            acc[0][1] = __builtin_amdgcn_wmma_f32_16x16x4_f32(false, a0, false, b1,
                            (short)0, acc[0][1], false, false);
            acc[1][0] = __builtin_amdgcn_wmma_f32_16x16x4_f32(false, a1, false, b0,
                            (short)0, acc[1][0], false, false);
            acc[1][1] = __builtin_amdgcn_wmma_f32_16x16x4_f32(false, a1, false, b1,
                            (short)0, acc[1][1], false, false);
        }
    };

    // ---- software pipeline: fill(it+1) overlaps compute(it) ----
    stage(0, 0);
    WAIT_ASYNC();
    __syncthreads();

    #pragma unroll 2
    for (int it = 0; it < N_ITER; ++it) {
        const int cur = it & 1;
        if (it + 1 < N_ITER)
            stage(it + 1, cur ^ 1);       // async fill of the other buffer
        compute(cur);                     // WMMA on the current buffer
        WAIT_ASYNC();                     // own fills landed
        __syncthreads();                  // everyone's fills landed, reads retired
    }

    // ---- epilogue: lanes sweep pixels (contiguous hw) -> coalesced stores ----
    #pragma unroll
    for (int ni = 0; ni < 2; ++ni) {
        const int p  = p_tile + wn + ni * 16 + l16;
        const int pn = p / HW;
        const int pr = p - pn * HW;
        float* out_base = out + (size_t)pn * K_OUT * HW + pr;
        #pragma unroll
        for (int mi = 0; mi < 2; ++mi) {
            #pragma unroll
            for (int i = 0; i < 8; ++i) {
                const int o = m_tile + wm + mi * 16 + half * 8 + i;
                out_base[(size_t)o * HW] = acc[mi][ni][i] + bias[o];
            }
        }
    }
}

extern "C" void kernel_launch(void* const* d_in, const int* in_sizes, int n_in,
                              void* d_out, int out_size, void* d_ws, size_t ws_size,
                              hipStream_t stream) {
    (void)in_sizes; (void)n_in; (void)out_size; (void)d_ws; (void)ws_size;
    const float* x    = (const float*)d_in[0];
    const float* wgt  = (const float*)d_in[1];
    const float* bias = (const float*)d_in[2];
    float* out        = (float*)d_out;

    dim3 grid((N_IMGS * HW) / TILE_N, K_OUT / TILE_M);  // (784, 4)
    dim3 block(256);
    hipLaunchKernelGGL(conv3x3_wmma_f32, grid, block, 0, stream, x, wgt, bias, out);
}